// ChebyshevEncoder_34557306864106
// MI455X (gfx1250) — compile-verified
//
#include <hip/hip_runtime.h>

// Problem constants (from reference)
#define BATCH   4096
#define INSZ    512
#define MT      8        // max_terms (= kernel_size)
#define NH      4        // heads
#define KS      8
#define FLAT    (NH * INSZ * KS)   // 16384 features per row
#define ROWS    16       // batch rows per block  (WMMA M dim)
#define LN_EPS  1e-5f

typedef float v2f __attribute__((ext_vector_type(2)));
typedef float v8f __attribute__((ext_vector_type(8)));

// Fast SiLU: t * sigmoid(t) with hardware v_rcp_f32 (1-ulp).
__device__ __forceinline__ float silu_fast(float t) {
    return t * __builtin_amdgcn_rcpf(1.0f + __expf(-t));
}

// One sweep over this wave's 32 feature pairs x 4 heads.
// STORE=false: accumulate per-row sum/sumsq.  STORE=true: normalize + stream out.
// Feature pair (i0,i1) packed into N columns 0-7 / 8-15 of one C tile via
// per-feature A fragments and complementary zero-masked B halves -> every
// lane/element of C is valid: no exec-mask branches in the hot loop.
template <bool STORE>
__device__ __forceinline__ void sweep(
    const float* __restrict__ xrow,    // x + (base+lm)*INSZ : this lane's A row
    const float* __restrict__ scale,
    const float* __restrict__ pw,
    const float* __restrict__ kern,
    const float* __restrict__ gamma,
    const float* __restrict__ beta,
    float* __restrict__ out,
    int wave, int half, bool lowcol, int kcol, int base,
    float* __restrict__ accs, float* __restrict__ accq,
    const float* __restrict__ mr, const float* __restrict__ rs)
{
    for (int ii = 0; ii < 32; ++ii) {
        const int i0 = wave * 64 + 2 * ii;
        const int i1 = i0 + 1;
        const int isel = lowcol ? i0 : i1;   // feature owned by this lane's N column

        // Chebyshev recurrences (row base+lm) for both features, in registers.
        float Ta[MT], Tb[MT];
        {
            const float xa = xrow[i0] * scale[i0];
            Ta[0] = 1.0f; Ta[1] = xa;
            #pragma unroll
            for (int m = 2; m < MT; ++m) Ta[m] = 2.0f * xa * Ta[m - 1] - Ta[m - 2];
            const float xb = xrow[i1] * scale[i1];
            Tb[0] = 1.0f; Tb[1] = xb;
            #pragma unroll
            for (int m = 2; m < MT; ++m) Tb[m] = 2.0f * xb * Tb[m - 1] - Tb[m - 2];
        }

        // A fragments (16x4 f32): lanes 0-15 -> K rows 0,1 ; lanes 16-31 -> 2,3
        v2f a00, a01, a10, a11;
        a00.x = half ? Ta[2] : Ta[0];  a00.y = half ? Ta[3] : Ta[1];
        a01.x = half ? Ta[6] : Ta[4];  a01.y = half ? Ta[7] : Ta[5];
        a10.x = half ? Tb[2] : Tb[0];  a10.y = half ? Tb[3] : Tb[1];
        a11.x = half ? Tb[6] : Tb[4];  a11.y = half ? Tb[7] : Tb[5];

        for (int h = 0; h < NH; ++h) {
            const int hi = h * INSZ + isel;
            const float* kb = kern + (size_t)hi * MT * KS;   // [m][k], lane's feature

            // Kernel values for this lane's column (4 m-rows for B V0/V1, 2 K-groups)
            const float kv0 = kb[(0 + 2 * half) * KS + kcol];
            const float kv1 = kb[(1 + 2 * half) * KS + kcol];
            const float kv4 = kb[(4 + 2 * half) * KS + kcol];
            const float kv5 = kb[(5 + 2 * half) * KS + kcol];

            // C accumulates 4 WMMAs: feature i0 into cols 0-7, i1 into cols 8-15.
            v2f b;
            v8f c = {};
            b.x = lowcol ? kv0 : 0.0f;  b.y = lowcol ? kv1 : 0.0f;
            c = __builtin_amdgcn_wmma_f32_16x16x4_f32(false, a00, false, b,
                                                      (short)0, c, false, false);
            b.x = lowcol ? kv4 : 0.0f;  b.y = lowcol ? kv5 : 0.0f;
            c = __builtin_amdgcn_wmma_f32_16x16x4_f32(false, a01, false, b,
                                                      (short)0, c, false, false);
            b.x = lowcol ? 0.0f : kv0;  b.y = lowcol ? 0.0f : kv1;
            c = __builtin_amdgcn_wmma_f32_16x16x4_f32(false, a10, false, b,
                                                      (short)0, c, false, false);
            b.x = lowcol ? 0.0f : kv4;  b.y = lowcol ? 0.0f : kv5;
            c = __builtin_amdgcn_wmma_f32_16x16x4_f32(false, a11, false, b,
                                                      (short)0, c, false, false);

            const float p = pw[hi * MT + kcol];

            if (STORE) {
                const int   fidx = hi * KS + kcol;   // flat feature index
                const float g  = gamma[fidx];
                const float bt = beta[fidx];
                #pragma unroll
                for (int v = 0; v < 8; ++v) {
                    const float s = silu_fast(c[v] * p);
                    const float o = (s - mr[v]) * rs[v] * g + bt;
                    // write-once 256 MB stream: non-temporal, keep L2 for kern/x
                    __builtin_nontemporal_store(
                        o, &out[(size_t)(base + v + 8 * half) * FLAT + fidx]);
                }
            } else {
                #pragma unroll
                for (int v = 0; v < 8; ++v) {
                    const float s = silu_fast(c[v] * p);
                    accs[v] += s;
                    accq[v] += s * s;
                }
            }
        }
    }
}

// One block = 16 batch rows (WMMA M dim). Two compute passes (recompute instead
// of spilling 1 MB of intermediates): pass 0 accumulates LayerNorm statistics,
// pass 1 recomputes and writes the single HBM output pass (roofline-optimal:
// 256 MB written once at 23.3 TB/s ~ 11 us floor; compute ~2 GFLOP is noise).
__global__ __launch_bounds__(256) void cheb_encoder_kernel(
    const float* __restrict__ x,       // [BATCH][INSZ]
    const float* __restrict__ scale,   // [INSZ]
    const float* __restrict__ pw,      // [NH][INSZ][MT]
    const float* __restrict__ kern,    // [NH][INSZ][MT][KS]
    const float* __restrict__ gamma,   // [FLAT]
    const float* __restrict__ beta,    // [FLAT]
    float* __restrict__ out)           // [BATCH][FLAT]
{
    __shared__ float s_sum[ROWS];
    __shared__ float s_sq[ROWS];
    __shared__ float s_mean[ROWS];
    __shared__ float s_rstd[ROWS];

    const int tid  = threadIdx.x;
    const int lane = tid & 31;
    const int wave = tid >> 5;        // 8 waves; wave owns features [wave*64, wave*64+64)
    const int half = lane >> 4;       // half-wave select (K rows for A/B, +8 rows for C)
    const int lm   = lane & 15;       // A: M row loaded by this lane; B/C: N column
    const int kcol = lane & 7;        // k index within this lane's feature
    const bool lowcol = (lm < 8);     // N column group: feature i0 (cols 0-7) vs i1
    const int base = blockIdx.x * ROWS;

    if (tid < ROWS) { s_sum[tid] = 0.0f; s_sq[tid] = 0.0f; }
    __syncthreads();

    const float* xrow = x + (size_t)(base + lm) * INSZ;   // this lane's A-matrix row

    float accs[8], accq[8];
    #pragma unroll
    for (int v = 0; v < 8; ++v) { accs[v] = 0.0f; accq[v] = 0.0f; }

    // Pass 0: statistics
    sweep<false>(xrow, scale, pw, kern, gamma, beta, out,
                 wave, half, lowcol, kcol, base, accs, accq, nullptr, nullptr);

    // Block-local LayerNorm reduction: every lane's accumulators are valid.
    #pragma unroll
    for (int v = 0; v < 8; ++v) {
        const int r = v + 8 * half;
        atomicAdd(&s_sum[r], accs[v]);   // ds_add_f32
        atomicAdd(&s_sq[r],  accq[v]);
    }
    __syncthreads();
    if (tid < ROWS) {
        const float mean = s_sum[tid] * (1.0f / (float)FLAT);
        const float var  = s_sq[tid]  * (1.0f / (float)FLAT) - mean * mean;
        s_mean[tid] = mean;
        s_rstd[tid] = rsqrtf(var + LN_EPS);
    }
    __syncthreads();

    float mr[8], rs[8];
    #pragma unroll
    for (int v = 0; v < 8; ++v) {
        mr[v] = s_mean[v + 8 * half];
        rs[v] = s_rstd[v + 8 * half];
    }

    // Pass 1: recompute, normalize, stream out
    sweep<true>(xrow, scale, pw, kern, gamma, beta, out,
                wave, half, lowcol, kcol, base, nullptr, nullptr, mr, rs);
}

extern "C" void kernel_launch(void* const* d_in, const int* in_sizes, int n_in,
                              void* d_out, int out_size, void* d_ws, size_t ws_size,
                              hipStream_t stream) {
    (void)in_sizes; (void)n_in; (void)d_ws; (void)ws_size; (void)out_size;
    const float* x     = (const float*)d_in[0];
    const float* scale = (const float*)d_in[1];
    const float* pw    = (const float*)d_in[2];
    const float* kern  = (const float*)d_in[3];
    const float* gamma = (const float*)d_in[4];
    const float* beta  = (const float*)d_in[5];
    float* out = (float*)d_out;

    dim3 grid(BATCH / ROWS);   // 256 blocks of 16 batch rows
    dim3 block(256);           // 8 wave32 waves; wave w owns features [64w, 64w+64)
    cheb_encoder_kernel<<<grid, block, 0, stream>>>(x, scale, pw, kern, gamma, beta, out);
}